// ScatteringClassifier_51573967290687
// MI455X (gfx1250) — compile-verified
//
#include <hip/hip_runtime.h>
#include <math.h>

typedef float v2f __attribute__((ext_vector_type(2)));
typedef float v8f __attribute__((ext_vector_type(8)));

#define T_DIM 200
#define T_PAD 208          // pad N to multiple of 16
#define P_DIM 784          // 49 * 16
#define B_DIM 256          // 16 * 16
#define K2    400          // [a | b] inner dim for G-gemm

// workspace layout (float offsets)
#define OFF_CR   0
#define OFF_CI   256
#define OFF_AMAT 512                          // 784 x 400
#define OFF_KMAT (OFF_AMAT + P_DIM * K2)      // 400 x 208
#define OFF_G    (OFF_KMAT + K2 * T_PAD)      // 784 x 208
#define OFF_XV   (OFF_G + P_DIM * T_PAD)      // 256 x 784
#define OFF_R    (OFF_XV + B_DIM * P_DIM)     // 256 x 208
// total = 814,336 floats ~= 3.26 MB

#define TWO_PI 6.28318530717958647692f

// c[k] = FFT(waveform)[k] * envelope[k]   (200-pt direct DFT, one block)
__global__ void prep_c_kernel(const float* __restrict__ wfr, const float* __restrict__ wfi,
                              const float* __restrict__ envr, const float* __restrict__ envi,
                              float* __restrict__ ws) {
    int k = threadIdx.x;
    if (k >= T_DIM) return;
    float wr = 0.f, wi = 0.f;
    for (int t = 0; t < T_DIM; ++t) {
        int m = (k * t) % T_DIM;                  // keep sincos argument small
        float s, c;
        __sincosf(-TWO_PI * (float)m / (float)T_DIM, &s, &c);
        float xr = wfr[t], xi = wfi[t];
        wr += xr * c - xi * s;
        wi += xr * s + xi * c;
    }
    float er = envr[k], ei = envi[k];
    ws[OFF_CR + k] = wr * er - wi * ei;
    ws[OFF_CI + k] = wr * ei + wi * er;
}

// Amat[p, k]       = Re(weights[p,k] * c[k])
// Amat[p, 200 + k] = Im(weights[p,k] * c[k])
__global__ void build_amat_kernel(const float* __restrict__ sr, const float* __restrict__ si,
                                  float* __restrict__ ws) {
    int idx = blockIdx.x * blockDim.x + threadIdx.x;
    if (idx >= P_DIM * T_DIM) return;
    int p = idx / T_DIM, k = idx - p * T_DIM;
    float cr = ws[OFF_CR + k], ci = ws[OFF_CI + k];
    float a = sr[idx], b = si[idx];
    ws[OFF_AMAT + p * K2 + k]         = a * cr - b * ci;
    ws[OFF_AMAT + p * K2 + T_DIM + k] = a * ci + b * cr;
}

// Kmat[k, t]       =  cos(2*pi*k*t/T)/T   (k < 200)
// Kmat[200+k, t]   = -sin(2*pi*k*t/T)/T
// columns 200..207 zero-padded
__global__ void build_kmat_kernel(float* __restrict__ ws) {
    int idx = blockIdx.x * blockDim.x + threadIdx.x;
    if (idx >= K2 * T_PAD) return;
    int k = idx / T_PAD, t = idx - k * T_PAD;
    float val = 0.f;
    if (t < T_DIM) {
        int kk = (k < T_DIM) ? k : (k - T_DIM);
        int m = (kk * t) % T_DIM;
        float s, c;
        __sincosf(TWO_PI * (float)m / (float)T_DIM, &s, &c);
        val = ((k < T_DIM) ? c : -s) * (1.0f / (float)T_DIM);
    }
    ws[OFF_KMAT + idx] = val;
}

// xv = x_batch * v  (broadcast over batch)
__global__ void build_xv_kernel(const float* __restrict__ x, const float* __restrict__ v,
                                float* __restrict__ ws) {
    int idx = blockIdx.x * blockDim.x + threadIdx.x;
    if (idx >= B_DIM * P_DIM) return;
    ws[OFF_XV + idx] = x[idx] * v[idx % P_DIM];
}

// f32 WMMA GEMM: C[M,N] = A[M,K] @ B[K,N], row-major, one wave per 16x16 tile.
// M % 16 == 0, N % 16 == 0, K % 4 == 0. Block = 32 threads (one wave32), no
// divergence anywhere so EXEC is all-ones as WMMA requires.
__global__ void wmma_gemm_f32(const float* __restrict__ A, const float* __restrict__ Bm,
                              float* __restrict__ C, int M, int N, int K) {
    int lane = threadIdx.x & 31;
    int n0 = blockIdx.x * 16;
    int m0 = blockIdx.y * 16;

    int rc    = lane & 15;          // row within A tile / col within B,C tile
    int khalf = (lane >> 4) * 2;    // lanes 0-15 hold K=0,1; lanes 16-31 hold K=2,3

    const float* aptr = A + (size_t)(m0 + rc) * K + khalf;   // a.x,a.y contiguous
    const float* bptr = Bm + (size_t)khalf * N + (n0 + rc);  // b.y is next K row

    v8f acc = {0.f, 0.f, 0.f, 0.f, 0.f, 0.f, 0.f, 0.f};
    for (int kk = 0; kk < K; kk += 4) {
        v2f a, b;
        a.x = aptr[0];
        a.y = aptr[1];
        b.x = bptr[0];
        b.y = bptr[N];
        // (neg_a, A, neg_b, B, c_mod, C, reuse_a, reuse_b)
        acc = __builtin_amdgcn_wmma_f32_16x16x4_f32(false, a, false, b,
                                                    (short)0, acc, false, false);
        aptr += 4;
        bptr += 4 * (size_t)N;
    }

    // C/D layout: VGPR r -> row m0 + r (+8 for lanes 16-31), col n0 + (lane&15)
    int rbase = m0 + ((lane >> 4) * 8);
    float* cptr = C + (size_t)rbase * N + (n0 + rc);
#pragma unroll
    for (int r = 0; r < 8; ++r) {
        cptr[(size_t)r * N] = acc[r];
    }
}

// chunk-max(|.|) via sqrt(x^2+eps) over 10 chunks of 20, then 10x10 linear
__global__ void finalize_kernel(const float* __restrict__ lin_w, const float* __restrict__ lin_b,
                                const float* __restrict__ ws_r, float* __restrict__ out) {
    int b = blockIdx.x * blockDim.x + threadIdx.x;
    if (b >= B_DIM) return;
    const float* rrow = ws_r + (size_t)b * T_PAD;
    float cm[10];
#pragma unroll
    for (int c = 0; c < 10; ++c) {
        float mx = 0.f;   // sqrt(x^2 + eps) > 0 always
        for (int t = 0; t < 20; ++t) {
            float x = rrow[c * 20 + t];
            mx = fmaxf(mx, sqrtf(x * x + 1e-20f));
        }
        cm[c] = mx;
    }
#pragma unroll
    for (int j = 0; j < 10; ++j) {
        float s = lin_b[j];
#pragma unroll
        for (int c = 0; c < 10; ++c) s += cm[c] * lin_w[j * 10 + c];
        out[b * 10 + j] = s;
    }
}

extern "C" void kernel_launch(void* const* d_in, const int* in_sizes, int n_in,
                              void* d_out, int out_size, void* d_ws, size_t ws_size,
                              hipStream_t stream) {
    const float* x   = (const float*)d_in[0];   // [256, 784]
    const float* v   = (const float*)d_in[1];   // [784]
    const float* sr  = (const float*)d_in[2];   // [784, 200]
    const float* si  = (const float*)d_in[3];   // [784, 200]
    const float* er  = (const float*)d_in[4];   // [200]
    const float* ei  = (const float*)d_in[5];   // [200]
    const float* wfr = (const float*)d_in[6];   // [200]
    const float* wfi = (const float*)d_in[7];   // [200]
    const float* lw  = (const float*)d_in[8];   // [10, 10]
    const float* lb  = (const float*)d_in[9];   // [10]
    float* ws  = (float*)d_ws;
    float* out = (float*)d_out;

    prep_c_kernel<<<1, 256, 0, stream>>>(wfr, wfi, er, ei, ws);
    build_amat_kernel<<<(P_DIM * T_DIM + 255) / 256, 256, 0, stream>>>(sr, si, ws);
    build_kmat_kernel<<<(K2 * T_PAD + 255) / 256, 256, 0, stream>>>(ws);
    build_xv_kernel<<<(B_DIM * P_DIM + 255) / 256, 256, 0, stream>>>(x, v, ws);

    // G[784, 208] = Amat[784, 400] @ Kmat[400, 208]
    dim3 g1(T_PAD / 16, P_DIM / 16);
    wmma_gemm_f32<<<g1, 32, 0, stream>>>(ws + OFF_AMAT, ws + OFF_KMAT, ws + OFF_G,
                                         P_DIM, T_PAD, K2);
    // r[256, 208] = xv[256, 784] @ G[784, 208]
    dim3 g2(T_PAD / 16, B_DIM / 16);
    wmma_gemm_f32<<<g2, 32, 0, stream>>>(ws + OFF_XV, ws + OFF_G, ws + OFF_R,
                                         B_DIM, T_PAD, P_DIM);

    finalize_kernel<<<(B_DIM + 63) / 64, 64, 0, stream>>>(lw, lb, ws + OFF_R, out);
}